// GraphTrans_70832600646276
// MI455X (gfx1250) — compile-verified
//
#include <hip/hip_runtime.h>
#include <math.h>

// Shapes (fixed by the reference)
#define BB   8
#define TT   96
#define SS   192
#define DD   512          // D_IN == D_MEM == 512
#define MQ   (BB*TT)      // 768
#define MU   (BB*SS)      // 1536

typedef __attribute__((ext_vector_type(2))) float v2f;
typedef __attribute__((ext_vector_type(8))) float v8f;

__device__ __forceinline__ v8f wmma_f32_16x16x4(v2f a, v2f b, v8f c) {
  // 8 args: (neg_a, A, neg_b, B, c_mod, C, reuse_a, reuse_b)
  return __builtin_amdgcn_wmma_f32_16x16x4_f32(false, a, false, b, (short)0, c,
                                               false, false);
}

// C[M,N] = A[M,K](row-major) * B[N,K](row-major)^T (+ bias[N]); one wave per
// 16x16 tile. Both operands are K-contiguous -> float2 loads per lane.
// WMMA f32 16x16x4 operand layout: lanes 0-15 hold K=k,k+1; lanes 16-31 K=k+2,k+3.
__global__ __launch_bounds__(256)
void gemm_nt_wmma(const float* __restrict__ A, int lda,
                  const float* __restrict__ B, int ldb,
                  const float* __restrict__ bias,
                  float* __restrict__ C, int ldc,
                  int M, int N, int K) {
  const int wavesPerBlock = blockDim.x >> 5;
  const int gw   = blockIdx.x * wavesPerBlock + (threadIdx.x >> 5);
  const int lane = threadIdx.x & 31;
  const int mTiles = M >> 4, nTiles = N >> 4;
  if (gw >= mTiles * nTiles) return;               // wave-uniform: EXEC stays all-1s
  const int mT = gw / nTiles, nT = gw % nTiles;
  const int row0 = mT << 4, col0 = nT << 4;
  const int half = lane >> 4, idx = lane & 15;

  const float* Arow = A + (size_t)(row0 + idx) * lda;
  const float* Brow = B + (size_t)(col0 + idx) * ldb;
  v8f acc = {};
  for (int k = 0; k < K; k += 4) {
    const int kk = k + 2 * half;
    v2f a, b;
    a.x = Arow[kk]; a.y = Arow[kk + 1];
    b.x = Brow[kk]; b.y = Brow[kk + 1];
    acc = wmma_f32_16x16x4(a, b, acc);
  }
  const float bb = bias ? bias[col0 + idx] : 0.0f;
  #pragma unroll
  for (int r = 0; r < 8; ++r) {
    const int row = row0 + r + 8 * half;           // C/D layout: VGPR r -> M=r / M=r+8
    C[(size_t)row * ldc + col0 + idx] = acc[r] + bb;
  }
}

// Per-(b,t) additive score + masked softmax.
// align[b,t,s] = sum_m tanh(wq[b,t,m] + uh[b,s,m]) * v[m]; softmax over s<len.
__global__ __launch_bounds__(256)
void score_softmax(const float* __restrict__ wq,     // [B*T, D]
                   const float* __restrict__ uh,     // [B*S, D]
                   const float* __restrict__ v,      // [D]
                   const int*   __restrict__ mem_masks, // [B]
                   float* __restrict__ align_out) {  // [B*T, S]
  const int bt = blockIdx.x;
  const int b  = bt / TT;
  __shared__ float s_wq[DD];
  __shared__ float s_v[DD];
  __shared__ float s_score[SS];
  __shared__ float s_red[32];

  const float* wqrow = wq + (size_t)bt * DD;
  for (int i = threadIdx.x; i < DD; i += blockDim.x) {
    s_wq[i] = wqrow[i];
    s_v[i]  = v[i];
  }
  __syncthreads();

  const int waveId = threadIdx.x >> 5;
  const int lane   = threadIdx.x & 31;
  const int nWaves = blockDim.x >> 5;
  const float* uhB = uh + (size_t)b * SS * DD;

  for (int s = waveId; s < SS; s += nWaves) {
    const float* uhrow = uhB + (size_t)s * DD;
    float sum = 0.0f;
    for (int m = lane; m < DD; m += 32)
      sum += tanhf(s_wq[m] + uhrow[m]) * s_v[m];
    #pragma unroll
    for (int off = 16; off >= 1; off >>= 1)
      sum += __shfl_xor(sum, off, 32);
    if (lane == 0) s_score[s] = sum;
  }
  __syncthreads();

  const int len = mem_masks[b];

  // masked max
  float mx = -INFINITY;
  for (int s = threadIdx.x; s < SS; s += blockDim.x)
    if (s < len) mx = fmaxf(mx, s_score[s]);
  #pragma unroll
  for (int off = 16; off >= 1; off >>= 1)
    mx = fmaxf(mx, __shfl_xor(mx, off, 32));
  if (lane == 0) s_red[waveId] = mx;
  __syncthreads();
  if (threadIdx.x < 32) {
    float m2 = (threadIdx.x < (unsigned)nWaves) ? s_red[threadIdx.x] : -INFINITY;
    #pragma unroll
    for (int off = 16; off >= 1; off >>= 1)
      m2 = fmaxf(m2, __shfl_xor(m2, off, 32));
    if (threadIdx.x == 0) s_red[0] = m2;
  }
  __syncthreads();
  mx = s_red[0];
  __syncthreads();

  // exp + sum
  float sum = 0.0f;
  for (int s = threadIdx.x; s < SS; s += blockDim.x) {
    const float e = (s < len) ? __expf(s_score[s] - mx) : 0.0f;
    s_score[s] = e;
    sum += e;
  }
  #pragma unroll
  for (int off = 16; off >= 1; off >>= 1)
    sum += __shfl_xor(sum, off, 32);
  if (lane == 0) s_red[waveId] = sum;
  __syncthreads();
  if (threadIdx.x < 32) {
    float s2 = (threadIdx.x < (unsigned)nWaves) ? s_red[threadIdx.x] : 0.0f;
    #pragma unroll
    for (int off = 16; off >= 1; off >>= 1)
      s2 += __shfl_xor(s2, off, 32);
    if (threadIdx.x == 0) s_red[0] = s2;
  }
  __syncthreads();
  const float inv = 1.0f / s_red[0];
  float* out = align_out + (size_t)bt * SS;
  for (int s = threadIdx.x; s < SS; s += blockDim.x)
    out[s] = s_score[s] * inv;
}

// c[b,t,d] = sum_s P[b,t,s] * mems[b,s,d]  (batched NN GEMM, M=T,N=D,K=S)
__global__ __launch_bounds__(256)
void ctx_gemm_wmma(const float* __restrict__ P,     // [B,T,S]
                   const float* __restrict__ Mems,  // [B,S,D]
                   float* __restrict__ C) {         // [B,T,D]
  const int wavesPerBlock = blockDim.x >> 5;
  const int gw   = blockIdx.x * wavesPerBlock + (threadIdx.x >> 5);
  const int lane = threadIdx.x & 31;
  const int mTiles = TT >> 4, nTiles = DD >> 4;
  const int perB = mTiles * nTiles;
  if (gw >= BB * perB) return;
  const int b = gw / perB;
  const int tile = gw % perB;
  const int mT = tile / nTiles, nT = tile % nTiles;
  const int row0 = mT << 4, col0 = nT << 4;
  const int half = lane >> 4, idx = lane & 15;

  const float* Arow = P + (size_t)b * TT * SS + (size_t)(row0 + idx) * SS;
  const float* Bb   = Mems + (size_t)b * SS * DD;
  v8f acc = {};
  for (int k = 0; k < SS; k += 4) {
    const int kk = k + 2 * half;
    v2f a, bv;
    a.x  = Arow[kk];
    a.y  = Arow[kk + 1];
    bv.x = Bb[(size_t)kk * DD + col0 + idx];
    bv.y = Bb[(size_t)(kk + 1) * DD + col0 + idx];
    acc = wmma_f32_16x16x4(a, bv, acc);
  }
  float* Cb = C + (size_t)b * TT * DD;
  #pragma unroll
  for (int r = 0; r < 8; ++r)
    Cb[(size_t)(row0 + r + 8 * half) * DD + col0 + idx] = acc[r];
}

// attn_h = [c, inputs] @ Wout^T + bout, without materializing the concat.
// M=768, N=512, K=1024; k<512 pulls from c, k>=512 from inputs.
__global__ __launch_bounds__(256)
void out_gemm_wmma(const float* __restrict__ Ctx,   // [M,512]
                   const float* __restrict__ In,    // [M,512]
                   const float* __restrict__ Wout,  // [512,1024] row-major
                   const float* __restrict__ bout,  // [512]
                   float* __restrict__ Out) {       // [M,512]
  const int wavesPerBlock = blockDim.x >> 5;
  const int gw   = blockIdx.x * wavesPerBlock + (threadIdx.x >> 5);
  const int lane = threadIdx.x & 31;
  const int mTiles = MQ >> 4, nTiles = DD >> 4;
  if (gw >= mTiles * nTiles) return;
  const int mT = gw / nTiles, nT = gw % nTiles;
  const int row0 = mT << 4, col0 = nT << 4;
  const int half = lane >> 4, idx = lane & 15;

  const float* Crow = Ctx + (size_t)(row0 + idx) * DD;
  const float* Irow = In  + (size_t)(row0 + idx) * DD;
  const float* Brow = Wout + (size_t)(col0 + idx) * (2 * DD);
  v8f acc = {};
  for (int k = 0; k < 2 * DD; k += 4) {
    const float* Arow = (k < DD) ? (Crow + k) : (Irow + (k - DD));
    const int o = 2 * half;
    v2f a, bv;
    a.x  = Arow[o];
    a.y  = Arow[o + 1];
    bv.x = Brow[k + o];
    bv.y = Brow[k + o + 1];
    acc = wmma_f32_16x16x4(a, bv, acc);
  }
  const float bb = bout[col0 + idx];
  #pragma unroll
  for (int r = 0; r < 8; ++r)
    Out[(size_t)(row0 + r + 8 * half) * DD + col0 + idx] = acc[r] + bb;
}

extern "C" void kernel_launch(void* const* d_in, const int* in_sizes, int n_in,
                              void* d_out, int out_size, void* d_ws, size_t ws_size,
                              hipStream_t stream) {
  const float* inputs    = (const float*)d_in[0];  // [B,T,512]
  const float* mems      = (const float*)d_in[1];  // [B,S,512]
  const int*   mem_masks = (const int*)  d_in[2];  // [B]
  const float* Wq        = (const float*)d_in[3];  // [512,512]
  const float* Wc        = (const float*)d_in[4];  // [512,512]
  const float* bc        = (const float*)d_in[5];  // [512]
  const float* v         = (const float*)d_in[6];  // [512]
  const float* Wout      = (const float*)d_in[7];  // [512,1024]
  const float* bout      = (const float*)d_in[8];  // [512]

  float* attn_h = (float*)d_out;                   // [B*T,512]
  float* align  = (float*)d_out + (size_t)MQ * DD; // [B*T,S]

  float* wq = (float*)d_ws;                        // [768,512]  (reused as c)
  float* uh = wq + (size_t)MQ * DD;                // [1536,512]

  const int threads = 256, wpb = threads / 32;

  // 1) wq = inputs @ Wq^T           (1536 tiles -> 192 blocks)
  gemm_nt_wmma<<<(MQ / 16) * (DD / 16) / wpb, threads, 0, stream>>>(
      inputs, DD, Wq, DD, nullptr, wq, DD, MQ, DD, DD);

  // 2) uh = mems @ Wc^T + bc        (3072 tiles -> 384 blocks)
  gemm_nt_wmma<<<(MU / 16) * (DD / 16) / wpb, threads, 0, stream>>>(
      mems, DD, Wc, DD, bc, uh, DD, MU, DD, DD);

  // 3) align = softmax(mask(v . tanh(wq+uh)))  -> directly into d_out
  score_softmax<<<MQ, threads, 0, stream>>>(wq, uh, v, mem_masks, align);

  // 4) c = align @ mems  (reuse wq scratch as c; 1536 tiles -> 192 blocks)
  ctx_gemm_wmma<<<BB * (TT / 16) * (DD / 16) / wpb, threads, 0, stream>>>(
      align, mems, wq);

  // 5) attn_h = [c, inputs] @ Wout^T + bout
  out_gemm_wmma<<<(MQ / 16) * (DD / 16) / wpb, threads, 0, stream>>>(
      wq, inputs, Wout, bout, attn_h);
}